// CrossNet_60352880443633
// MI455X (gfx1250) — compile-verified
//
#include <hip/hip_runtime.h>

typedef float v2f __attribute__((ext_vector_type(2)));
typedef float v4f __attribute__((ext_vector_type(4)));
typedef float v8f __attribute__((ext_vector_type(8)));

#define DIM            1024
#define ROWS_PER_WAVE  16
#define WAVES          4
#define BLOCK          (WAVES * 32)             // 128 threads, wave32
#define ROWS_PER_BLOCK (WAVES * ROWS_PER_WAVE)  // 64
#define CK             128                      // columns staged per chunk
#define XSTRIDE        132                      // 16B-aligned rows + conflict-free A-frag reads
#define WZPAD          132                      // zeroed tail of lds_w (>= CK + kp + 2)

// CrossNet collapsed: out = a3 * x0 + (b0+b1+b2),
//   a_{l+1} = a_l*(1 + x0.w_l) + e_l,  e_l = (sum_{i<l} b_i).w_l
__global__ __launch_bounds__(BLOCK) void crossnet_fused(
    const float* __restrict__ X,    // (B, DIM)
    const float* __restrict__ Wk,   // (3, DIM)
    const float* __restrict__ Bb,   // (3, DIM)
    float* __restrict__ Out)        // (B, DIM)
{
    __shared__ float lds_w[3 * DIM + WZPAD];               // staged kernels + zero pad
    __shared__ float lds_x[WAVES][ROWS_PER_WAVE * XSTRIDE];// per-wave x chunk (~33 KB)
    __shared__ float lds_b3[DIM];                          // b0+b1+b2
    __shared__ float lds_a[ROWS_PER_BLOCK];                // per-row scalar a3
    __shared__ float lds_red[2 * BLOCK];
    __shared__ float lds_e[2];                             // e1, e2

    const int tid = threadIdx.x;

    // ---- setup: stage W (+ zero pad), bias-sum, and e1/e2 constants ----
    float p1 = 0.f, p2 = 0.f;
    for (int j = tid; j < DIM; j += BLOCK) {
        float w0 = Wk[j], w1 = Wk[DIM + j], w2 = Wk[2 * DIM + j];
        lds_w[j] = w0; lds_w[DIM + j] = w1; lds_w[2 * DIM + j] = w2;
        float b0 = Bb[j], b1 = Bb[DIM + j], b2 = Bb[2 * DIM + j];
        lds_b3[j] = b0 + b1 + b2;
        p1 += b0 * w1;
        p2 += (b0 + b1) * w2;
    }
    for (int j = tid; j < WZPAD; j += BLOCK) lds_w[3 * DIM + j] = 0.f;
    lds_red[tid] = p1;
    lds_red[BLOCK + tid] = p2;
    __syncthreads();
    if (tid == 0) {
        float e1 = 0.f, e2 = 0.f;
        for (int i = 0; i < BLOCK; ++i) { e1 += lds_red[i]; e2 += lds_red[BLOCK + i]; }
        lds_e[0] = e1; lds_e[1] = e2;
    }
    __syncthreads();

    // ---- phase 1: per-wave 16-row tile, D = Xtile * [w0 w1 w2 0..] via WMMA ----
    const int wave = tid >> 5, lane = tid & 31;
    const int m  = lane & 15;            // A: row M / B: col N / C,D: col N
    const int kp = (lane >> 4) << 1;     // K pair offset: lanes 16-31 hold K+2,K+3
    float* myx = &lds_x[wave][0];
    const float* abase = &myx[m * XSTRIDE + kp];
    const int row0 = blockIdx.x * ROWS_PER_BLOCK + wave * ROWS_PER_WAVE;

    v8f acc = {};
    for (int kc = 0; kc < DIM; kc += CK) {
        // coalesced stage of 16 rows x CK cols into wave-private LDS
        #pragma unroll 4
        for (int r = 0; r < ROWS_PER_WAVE; ++r) {
            v4f v = *(const v4f*)&X[(size_t)(row0 + r) * DIM + kc + lane * 4];
            *(v4f*)&myx[r * XSTRIDE + lane * 4] = v;
        }
        // B base: lanes N<3 -> w_N[kc..], lanes N>=3 -> zeroed pad (branch-free inner loop)
        const float* wbase = (m < 3) ? &lds_w[m * DIM + kc + kp] : &lds_w[3 * DIM + kp];
        // 32 WMMA steps over the chunk (K advances by 4 per v_wmma_f32_16x16x4_f32)
        #pragma unroll
        for (int kk = 0; kk < CK; kk += 4) {
            v2f af = *(const v2f*)(abase + kk);   // A 16x4 fragment
            v2f bf = *(const v2f*)(wbase + kk);   // B 4x16 fragment (cols >=3 read zeros)
            acc = __builtin_amdgcn_wmma_f32_16x16x4_f32(
                false, af, false, bf, (short)0, acc, false, false);
        }
    }

    // ---- extract d0,d1,d2 per row (C/D layout: VGPR r -> M=r / r+8), recurrence ----
    #pragma unroll
    for (int r = 0; r < 8; ++r) {
        int mm = r + ((lane >> 4) << 3);
        myx[mm * 16 + m] = acc[r];
    }
    if (lane < 16) {
        float d0 = myx[lane * 16 + 0];
        float d1 = myx[lane * 16 + 1];
        float d2 = myx[lane * 16 + 2];
        float a = 1.f;
        a = a + a * d0;                       // e0 = 0
        a = a + (a * d1 + lds_e[0]);
        a = a + (a * d2 + lds_e[1]);
        lds_a[wave * ROWS_PER_WAVE + lane] = a;
    }
    __syncthreads();

    // ---- phase 2: out = a * x + b3 (x re-read hits L2; NT stores keep x resident) ----
    const int c0 = tid * (DIM / BLOCK);   // 8 contiguous cols per thread
    v4f bA = *(const v4f*)&lds_b3[c0];
    v4f bB = *(const v4f*)&lds_b3[c0 + 4];
    const int rowb = blockIdx.x * ROWS_PER_BLOCK;
    for (int r = 0; r < ROWS_PER_BLOCK; ++r) {
        size_t base = (size_t)(rowb + r) * DIM + c0;
        float av = lds_a[r];
        v4f xa = *(const v4f*)&X[base];
        v4f xb = *(const v4f*)&X[base + 4];
        v4f oa = xa * av + bA;
        v4f ob = xb * av + bB;
        __builtin_nontemporal_store(oa, (v4f*)&Out[base]);
        __builtin_nontemporal_store(ob, (v4f*)&Out[base + 4]);
    }
}

extern "C" void kernel_launch(void* const* d_in, const int* in_sizes, int n_in,
                              void* d_out, int out_size, void* d_ws, size_t ws_size,
                              hipStream_t stream) {
    const float* x  = (const float*)d_in[0];   // (16384, 1024) f32
    const float* wk = (const float*)d_in[1];   // (3, 1024) f32
    const float* bb = (const float*)d_in[2];   // (3, 1024) f32
    float* out = (float*)d_out;

    const int n_rows = in_sizes[0] / DIM;              // 16384
    const int grid   = n_rows / ROWS_PER_BLOCK;        // 256 blocks
    crossnet_fused<<<grid, BLOCK, 0, stream>>>(x, wk, bb, out);
}